// DA_rnn_48842368090440
// MI455X (gfx1250) — compile-verified
//
#include <hip/hip_runtime.h>
#include <cstddef>

// B=256, T=48, I=64, HE=HD=256
typedef float v2f __attribute__((ext_vector_type(2)));
typedef float v8f __attribute__((ext_vector_type(8)));

__device__ __forceinline__ v8f wmma_f32_k4(v2f a, v2f b, v8f c) {
  // V_WMMA_F32_16X16X4_F32 : D = A(16x4) x B(4x16) + C
  return __builtin_amdgcn_wmma_f32_16x16x4_f32(false, a, false, b, (short)0, c,
                                               false, false);
}

__device__ __forceinline__ float sigm(float x) { return 1.0f / (1.0f + expf(-x)); }

__device__ __forceinline__ float wave_sum(float s) {
  s += __shfl_xor(s, 16, 32);
  s += __shfl_xor(s, 8, 32);
  s += __shfl_xor(s, 4, 32);
  s += __shfl_xor(s, 2, 32);
  s += __shfl_xor(s, 1, 32);
  return s;
}
__device__ __forceinline__ float wave_max(float s) {
  s = fmaxf(s, __shfl_xor(s, 16, 32));
  s = fmaxf(s, __shfl_xor(s, 8, 32));
  s = fmaxf(s, __shfl_xor(s, 4, 32));
  s = fmaxf(s, __shfl_xor(s, 2, 32));
  s = fmaxf(s, __shfl_xor(s, 1, 32));
  return s;
}

// ---------------------------------------------------------------------------
// Kernel 1: per-(b,i) precompute. delta[B,T,I]; esc[B,T,I] = softmax logits
// (prefix_t Wa_x[s]*xn_s + sum_t Wa_m[t]*mask) -- h/c/ba terms cancel in softmax.
// ---------------------------------------------------------------------------
__global__ void prep_kernel(const float* __restrict__ X, const float* __restrict__ Xl,
                            const float* __restrict__ Wgx, const float* __restrict__ bgx,
                            const float* __restrict__ Wa,
                            float* __restrict__ delta, float* __restrict__ esc) {
  int idx = blockIdx.x * blockDim.x + threadIdx.x;
  if (idx >= 256 * 64) return;
  int b = idx >> 6, i = idx & 63;
  const float* xp  = X  + ((size_t)b * 48) * 64 + i;
  const float* xlp = Xl + ((size_t)b * 48) * 64 + i;
  float d = 0.f, cnt = 0.f, sum = 0.f, mc = 0.f;
  bool prev_obs = false;
  for (int t = 0; t < 48; ++t) {
    if (t > 0) d = prev_obs ? 1.f : d + 1.f;
    delta[((size_t)b * 48 + t) * 64 + i] = d;
    float x = xp[t * 64];
    bool obs = (x != -1.0f);
    prev_obs = obs;
    if (obs) { cnt += 1.f; sum += x; }
    mc += (obs ? 1.f : 0.f) * Wa[560 + t];  // Wa layout: [h(256)|c(256)|Xn(48)|mask(48)]
  }
  float xm = (cnt > 0.f) ? sum / cnt : 0.f;
  float wii = Wgx[i * 64 + i];  // FilterLinear keeps only the diagonal
  float bg = bgx[i];
  float pref = 0.f;
  for (int t = 0; t < 48; ++t) {
    float dv = delta[((size_t)b * 48 + t) * 64 + i];
    float dx = expf(-fmaxf(0.f, wii * dv + bg));
    float x = xp[t * 64];
    bool obs = (x != -1.0f);
    float xn = obs ? x : (dx * xlp[t * 64] + (1.f - dx) * xm);
    pref += Wa[512 + t] * xn;
    esc[((size_t)b * 48 + t) * 64 + i] = pref + mc;
  }
}

// ---------------------------------------------------------------------------
// Kernel 2 (generic WMMA GEMM): C[M,N] = act(A[M,K] @ W[N,ldw](cols koff..)^T + bias)
// One wave per 16x16 tile, fp32 WMMA K=4. act=1 -> exp(-relu(x)) (GRU-D dh).
// ---------------------------------------------------------------------------
__global__ __launch_bounds__(256) void gemm_wmma(const float* __restrict__ A, int M, int K,
                                                 const float* __restrict__ W, int ldw, int koff,
                                                 const float* __restrict__ bias,
                                                 float* __restrict__ C, int N, int act) {
  int wave = (int)((blockIdx.x * blockDim.x + threadIdx.x) >> 5);
  int lane = threadIdx.x & 31;
  int tilesN = N >> 4;
  int mt = wave / tilesN, nt = wave % tilesN;
  if (mt * 16 >= M) return;
  int col = lane & 15, kq = lane >> 4;
  const float* arow = A + ((size_t)(mt * 16 + col)) * K;        // M row for A-frag
  const float* wrow = W + ((size_t)(nt * 16 + col)) * ldw + koff; // N row of weights
  v8f acc;
  for (int j = 0; j < 8; ++j) acc[j] = 0.f;
  for (int k = 0; k < K; k += 4) {
    int klo = k + 2 * kq;
    float2 av = *(const float2*)(arow + klo);
    float2 bv = *(const float2*)(wrow + klo);
    v2f a;  a.x = av.x;  a.y = av.y;
    v2f bb; bb.x = bv.x; bb.y = bv.y;
    acc = wmma_f32_k4(a, bb, acc);
  }
  int n = nt * 16 + col;
  float bs = bias[n];
  for (int v = 0; v < 8; ++v) {
    int m = mt * 16 + kq * 8 + v;
    float val = acc[v] + bs;
    if (act == 1) val = expf(-fmaxf(0.f, val));
    C[(size_t)m * N + n] = val;
  }
}

// ---------------------------------------------------------------------------
// Kernel 3: row softmax over I=64 of esc, times X0 (missing zeroed) -> x~ in place
// ---------------------------------------------------------------------------
__global__ void alpha_kernel(float* __restrict__ esc, const float* __restrict__ X) {
  int row = blockIdx.x * blockDim.x + threadIdx.x;
  if (row >= 256 * 48) return;
  float* e = esc + (size_t)row * 64;
  const float* x = X + (size_t)row * 64;
  float m = -1e30f;
  for (int i = 0; i < 64; ++i) m = fmaxf(m, e[i]);
  float s = 0.f;
  for (int i = 0; i < 64; ++i) s += expf(e[i] - m);
  float inv = 1.f / s;
  for (int i = 0; i < 64; ++i) {
    float a = expf(e[i] - m) * inv;
    float xv = x[i];
    e[i] = a * ((xv != -1.0f) ? xv : 0.f);
  }
}

// ---------------------------------------------------------------------------
// Kernel 4: persistent encoder. Per block: 16 batch rows, all 48 LSTM steps.
// 16 waves; wave w owns hidden slice he=[16w,16w+16) for all 4 gates -> LSTM
// pointwise stays in the wave's own accumulators. h kept in LDS pre-scaled by
// the next step's GRU-D decay dh.
// ---------------------------------------------------------------------------
__global__ __launch_bounds__(512) void enc_kernel(const float* __restrict__ xt,
                                                  const float* __restrict__ dh,
                                                  const float* __restrict__ Whh,
                                                  const float* __restrict__ Wih,
                                                  const float* __restrict__ bih,
                                                  const float* __restrict__ bhh,
                                                  float* __restrict__ Xenc) {
  __shared__ float Ah[16][264];   // h (decayed), rows=batch-local
  __shared__ float Axt[16][72];   // x~ for current t
  int tid = threadIdx.x, lane = tid & 31, w = tid >> 5;
  int m0 = blockIdx.x * 16;
  int col = lane & 15, kq = lane >> 4;
  int he = w * 16 + col;
  for (int idx = tid; idx < 16 * 264; idx += 512) (&Ah[0][0])[idx] = 0.f;
  v8f cfr;
  for (int j = 0; j < 8; ++j) cfr[j] = 0.f;
  float bg0 = bih[he] + bhh[he];
  float bg1 = bih[256 + he] + bhh[256 + he];
  float bg2 = bih[512 + he] + bhh[512 + he];
  float bg3 = bih[768 + he] + bhh[768 + he];
  const float* wh0 = Whh + (size_t)he * 256;
  const float* wh1 = Whh + (size_t)(256 + he) * 256;
  const float* wh2 = Whh + (size_t)(512 + he) * 256;
  const float* wh3 = Whh + (size_t)(768 + he) * 256;
  const float* wi0 = Wih + (size_t)he * 64;
  const float* wi1 = Wih + (size_t)(256 + he) * 64;
  const float* wi2 = Wih + (size_t)(512 + he) * 64;
  const float* wi3 = Wih + (size_t)(768 + he) * 64;
  __syncthreads();
  for (int t = 0; t < 48; ++t) {
    for (int idx = tid; idx < 16 * 64; idx += 512) {
      int r = idx >> 6, k2 = idx & 63;
      Axt[r][k2] = xt[((size_t)(m0 + r) * 48 + t) * 64 + k2];
    }
    __syncthreads();
    v8f g0, g1, g2, g3;
    for (int j = 0; j < 8; ++j) { g0[j] = 0.f; g1[j] = 0.f; g2[j] = 0.f; g3[j] = 0.f; }
    for (int k = 0; k < 256; k += 4) {   // gates += (dh_t * h_{t-1}) @ Whh^T
      int klo = k + 2 * kq;
      v2f a; a.x = Ah[col][klo]; a.y = Ah[col][klo + 1];
      float2 b0 = *(const float2*)(wh0 + klo);
      float2 b1 = *(const float2*)(wh1 + klo);
      float2 b2 = *(const float2*)(wh2 + klo);
      float2 b3 = *(const float2*)(wh3 + klo);
      v2f vb;
      vb.x = b0.x; vb.y = b0.y; g0 = wmma_f32_k4(a, vb, g0);
      vb.x = b1.x; vb.y = b1.y; g1 = wmma_f32_k4(a, vb, g1);
      vb.x = b2.x; vb.y = b2.y; g2 = wmma_f32_k4(a, vb, g2);
      vb.x = b3.x; vb.y = b3.y; g3 = wmma_f32_k4(a, vb, g3);
    }
    for (int k = 0; k < 64; k += 4) {    // gates += x~_t @ Wih^T
      int klo = k + 2 * kq;
      v2f a; a.x = Axt[col][klo]; a.y = Axt[col][klo + 1];
      float2 b0 = *(const float2*)(wi0 + klo);
      float2 b1 = *(const float2*)(wi1 + klo);
      float2 b2 = *(const float2*)(wi2 + klo);
      float2 b3 = *(const float2*)(wi3 + klo);
      v2f vb;
      vb.x = b0.x; vb.y = b0.y; g0 = wmma_f32_k4(a, vb, g0);
      vb.x = b1.x; vb.y = b1.y; g1 = wmma_f32_k4(a, vb, g1);
      vb.x = b2.x; vb.y = b2.y; g2 = wmma_f32_k4(a, vb, g2);
      vb.x = b3.x; vb.y = b3.y; g3 = wmma_f32_k4(a, vb, g3);
    }
    __syncthreads();
    for (int v = 0; v < 8; ++v) {
      int m = kq * 8 + v, b = m0 + m;
      float ii = sigm(g0[v] + bg0);
      float ff = sigm(g1[v] + bg1);
      float gg = tanhf(g2[v] + bg2);
      float oo = sigm(g3[v] + bg3);
      float cn = ff * cfr[v] + ii * gg;
      cfr[v] = cn;
      float hn = oo * tanhf(cn);
      Xenc[((size_t)b * 48 + t) * 256 + he] = hn;
      float sc = (t < 47) ? dh[((size_t)b * 48 + t + 1) * 256 + he] : 0.f;
      Ah[m][he] = hn * sc;  // pre-decayed h for step t+1
    }
    __syncthreads();
  }
}

// ---------------------------------------------------------------------------
// Kernel 5: persistent decoder + final FC. Per block: 16 batch rows, 47 steps.
// Phase1: dcproj = d@Wd1_d^T + c@Wd1_c^T (WMMA).  Phase2: per-wave-per-row
// temporal attention (e, softmax over T, ctx, y~).  Phase3: decoder LSTM
// gates = d@dec_Whh^T (WMMA) + y~*dec_Wih + biases.  Then final two FCs.
// ---------------------------------------------------------------------------
__global__ __launch_bounds__(512) void dec_kernel(const float* __restrict__ P,
                                                  const float* __restrict__ Xenc,
                                                  const float* __restrict__ yprev,
                                                  const float* __restrict__ Wd1,
                                                  const float* __restrict__ Wd2,
                                                  const float* __restrict__ bd2,
                                                  const float* __restrict__ dWih,
                                                  const float* __restrict__ dWhh,
                                                  const float* __restrict__ bih,
                                                  const float* __restrict__ bhh,
                                                  const float* __restrict__ Wfc,
                                                  const float* __restrict__ bfc,
                                                  const float* __restrict__ Wf1,
                                                  const float* __restrict__ bf1,
                                                  const float* __restrict__ Wf2,
                                                  const float* __restrict__ bf2,
                                                  float* __restrict__ out) {
  __shared__ float Ad[16][264];   // d
  __shared__ float Ac[16][264];   // c
  __shared__ float Dc[16][264];   // dcproj, later reused as ctx
  __shared__ float ytil[16];
  int tid = threadIdx.x, lane = tid & 31, w = tid >> 5;
  int m0 = blockIdx.x * 16;
  int col = lane & 15, kq = lane >> 4;
  int he = w * 16 + col;
  for (int idx = tid; idx < 16 * 264; idx += 512) {
    (&Ad[0][0])[idx] = 0.f;
    (&Ac[0][0])[idx] = 0.f;
  }
  const float* wdd = Wd1 + (size_t)he * 768;        // d-part cols [0,256)
  const float* wdc = wdd + 256;                     // c-part cols [256,512)
  float bg0 = bih[he] + bhh[he];
  float bg1 = bih[256 + he] + bhh[256 + he];
  float bg2 = bih[512 + he] + bhh[512 + he];
  float bg3 = bih[768 + he] + bhh[768 + he];
  float wn0 = dWih[he], wn1 = dWih[256 + he], wn2 = dWih[512 + he], wn3 = dWih[768 + he];
  const float* wh0 = dWhh + (size_t)he * 256;
  const float* wh1 = dWhh + (size_t)(256 + he) * 256;
  const float* wh2 = dWhh + (size_t)(512 + he) * 256;
  const float* wh3 = dWhh + (size_t)(768 + he) * 256;
  int bw = m0 + w;  // this wave's batch row for phase 2 / final
  const float* Pb = P + (size_t)bw * 48 * 256;
  const float* Xb = Xenc + (size_t)bw * 48 * 256;
  __syncthreads();
  for (int t = 0; t < 47; ++t) {
    // ---- Phase 1: dcproj (bd1 already folded into P) ----
    v8f acc;
    for (int j = 0; j < 8; ++j) acc[j] = 0.f;
    for (int k = 0; k < 256; k += 4) {
      int klo = k + 2 * kq;
      v2f a; a.x = Ad[col][klo]; a.y = Ad[col][klo + 1];
      float2 bv = *(const float2*)(wdd + klo);
      v2f vb; vb.x = bv.x; vb.y = bv.y;
      acc = wmma_f32_k4(a, vb, acc);
    }
    for (int k = 0; k < 256; k += 4) {
      int klo = k + 2 * kq;
      v2f a; a.x = Ac[col][klo]; a.y = Ac[col][klo + 1];
      float2 bv = *(const float2*)(wdc + klo);
      v2f vb; vb.x = bv.x; vb.y = bv.y;
      acc = wmma_f32_k4(a, vb, acc);
    }
    for (int v = 0; v < 8; ++v) Dc[kq * 8 + v][he] = acc[v];
    __syncthreads();
    // ---- Phase 2: temporal attention, ctx, y~ (wave w <-> batch row bw) ----
    float e0 = -1e30f, e1 = -1e30f;
    for (int tp = 0; tp < 48; ++tp) {
      float s = 0.f;
      for (int h2 = lane; h2 < 256; h2 += 32)
        s += tanhf(Pb[tp * 256 + h2] + Dc[w][h2]) * Wd2[h2];
      s = wave_sum(s) + bd2[0];
      if (tp == lane) e0 = s;
      if (tp == lane + 32) e1 = s;
    }
    float mm = wave_max(fmaxf(e0, e1));
    float ss = wave_sum(expf(e0 - mm) + expf(e1 - mm));
    float inv = 1.f / ss;
    float beta0 = expf(e0 - mm) * inv, beta1 = expf(e1 - mm) * inv;
    float cxa[8];
    for (int j = 0; j < 8; ++j) cxa[j] = 0.f;
    for (int tp = 0; tp < 48; ++tp) {
      float bta = (tp < 32) ? __shfl(beta0, tp, 32) : __shfl(beta1, tp - 32, 32);
      for (int j = 0; j < 8; ++j) cxa[j] += bta * Xb[tp * 256 + lane + 32 * j];
    }
    float yt = 0.f;
    for (int j = 0; j < 8; ++j) {
      Dc[w][lane + 32 * j] = cxa[j];  // ctx overwrites this wave's dcproj row
      yt += cxa[j] * Wfc[lane + 32 * j];
    }
    yt = wave_sum(yt) + yprev[(size_t)bw * 47 + t] * Wfc[256] + bfc[0];
    if (lane == 0) ytil[w] = yt;
    __syncthreads();
    // ---- Phase 3: decoder LSTM ----
    v8f g0, g1, g2, g3;
    for (int j = 0; j < 8; ++j) { g0[j] = 0.f; g1[j] = 0.f; g2[j] = 0.f; g3[j] = 0.f; }
    for (int k = 0; k < 256; k += 4) {
      int klo = k + 2 * kq;
      v2f a; a.x = Ad[col][klo]; a.y = Ad[col][klo + 1];
      float2 b0 = *(const float2*)(wh0 + klo);
      float2 b1 = *(const float2*)(wh1 + klo);
      float2 b2 = *(const float2*)(wh2 + klo);
      float2 b3 = *(const float2*)(wh3 + klo);
      v2f vb;
      vb.x = b0.x; vb.y = b0.y; g0 = wmma_f32_k4(a, vb, g0);
      vb.x = b1.x; vb.y = b1.y; g1 = wmma_f32_k4(a, vb, g1);
      vb.x = b2.x; vb.y = b2.y; g2 = wmma_f32_k4(a, vb, g2);
      vb.x = b3.x; vb.y = b3.y; g3 = wmma_f32_k4(a, vb, g3);
    }
    __syncthreads();
    for (int v = 0; v < 8; ++v) {
      int m = kq * 8 + v;
      float ytv = ytil[m];
      float ii = sigm(g0[v] + ytv * wn0 + bg0);
      float ff = sigm(g1[v] + ytv * wn1 + bg1);
      float gg = tanhf(g2[v] + ytv * wn2 + bg2);
      float oo = sigm(g3[v] + ytv * wn3 + bg3);
      float cn = ff * Ac[m][he] + ii * gg;
      float dn = oo * tanhf(cn);
      Ac[m][he] = cn;
      Ad[m][he] = dn;
    }
    __syncthreads();
  }
  // ---- Final: out[b] = (concat(d, ctx) @ Wf1^T + bf1) @ Wf2^T + bf2 ----
  float accOut = 0.f;
  for (int j = 0; j < 256; ++j) {
    const float* wr = Wf1 + (size_t)j * 512;
    float s = 0.f;
    for (int k = lane; k < 256; k += 32) s += Ad[w][k] * wr[k] + Dc[w][k] * wr[256 + k];
    s = wave_sum(s) + bf1[j];
    accOut += s * Wf2[j];
  }
  if (lane == 0) out[bw] = accOut + bf2[0];
}

// ---------------------------------------------------------------------------
extern "C" void kernel_launch(void* const* d_in, const int* in_sizes, int n_in,
                              void* d_out, int out_size, void* d_ws, size_t ws_size,
                              hipStream_t stream) {
  const float* X     = (const float*)d_in[0];
  const float* Xlast = (const float*)d_in[1];
  const float* yprev = (const float*)d_in[2];
  const float* Wgx   = (const float*)d_in[3];
  const float* bgx   = (const float*)d_in[4];
  const float* Wgh   = (const float*)d_in[5];
  const float* bgh   = (const float*)d_in[6];
  const float* eWih  = (const float*)d_in[7];
  const float* eWhh  = (const float*)d_in[8];
  const float* ebih  = (const float*)d_in[9];
  const float* ebhh  = (const float*)d_in[10];
  const float* Wa    = (const float*)d_in[11];
  const float* Wd1   = (const float*)d_in[13];
  const float* bd1   = (const float*)d_in[14];
  const float* Wd2   = (const float*)d_in[15];
  const float* bd2   = (const float*)d_in[16];
  const float* dWih  = (const float*)d_in[17];
  const float* dWhh  = (const float*)d_in[18];
  const float* dbih  = (const float*)d_in[19];
  const float* dbhh  = (const float*)d_in[20];
  const float* Wfc   = (const float*)d_in[21];
  const float* bfc   = (const float*)d_in[22];
  const float* Wf1   = (const float*)d_in[23];
  const float* bf1   = (const float*)d_in[24];
  const float* Wf2   = (const float*)d_in[25];
  const float* bf2   = (const float*)d_in[26];

  float* ws = (float*)d_ws;
  float* delta = ws;                      // [12288,64]
  float* esc   = delta + 786432;          // [12288,64]  (logits -> x~ in place)
  float* dh    = esc + 786432;            // [12288,256]
  float* xenc  = dh + 3145728;            // [12288,256]
  float* Pbuf  = xenc + 3145728;          // [12288,256]

  prep_kernel<<<64, 256, 0, stream>>>(X, Xlast, Wgx, bgx, Wa, delta, esc);
  // dh = exp(-relu(delta @ Wgh^T + bgh))  [12288,64]x[64,256]
  gemm_wmma<<<1536, 256, 0, stream>>>(delta, 12288, 64, Wgh, 64, 0, bgh, dh, 256, 1);
  alpha_kernel<<<48, 256, 0, stream>>>(esc, X);
  enc_kernel<<<16, 512, 0, stream>>>(esc, dh, eWhh, eWih, ebih, ebhh, xenc);
  // P = X_enc @ Wd1_x^T + bd1   (Wd1 cols [512,768))
  gemm_wmma<<<1536, 256, 0, stream>>>(xenc, 12288, 256, Wd1, 768, 512, bd1, Pbuf, 256, 0);
  dec_kernel<<<16, 512, 0, stream>>>(Pbuf, xenc, yprev, Wd1, Wd2, bd2, dWih, dWhh,
                                     dbih, dbhh, Wfc, bfc, Wf1, bf1, Wf2, bf2,
                                     (float*)d_out);
}